// GATNet_28192165331190
// MI455X (gfx1250) — compile-verified
//
#include <hip/hip_runtime.h>
#include <math.h>

#define NNODES 50000
#define NEDGES 800000
#define ETOT   (NEDGES + NNODES)
#define INDIM  128
#define H1C    4
#define C1C    32
#define HID    128
#define OUTD   64
#define NGRAPH 64

typedef float v2f __attribute__((ext_vector_type(2)));
typedef float v8f __attribute__((ext_vector_type(8)));

// ---------------- f32 WMMA GEMM: C[M,Ncol] = A[M,K] @ B[K,Ncol] ----------------
// One wave per 16x16 output tile; K-loop of V_WMMA_F32_16X16X4_F32.
// A frag (16x4): VGPR0 = K0 (lanes 0-15) / K2 (lanes 16-31); VGPR1 = K1 / K3.
// B frag (4x16): VGPR0 = row K0 (lanes 0-15) / row K2 (lanes 16-31); VGPR1 = K1 / K3.
// C/D (16x16): VGPR v, lanes 0-15 -> M=v; lanes 16-31 -> M=v+8.
__global__ void gemm_wmma_f32(const float* __restrict__ A, const float* __restrict__ B,
                              float* __restrict__ Cm, int M, int K, int Ncol) {
  int gwave = (int)((blockIdx.x * blockDim.x + threadIdx.x) >> 5);
  int lane  = (int)(threadIdx.x & 31);
  int tilesN = Ncol >> 4;
  int numTiles = (M >> 4) * tilesN;
  if (gwave >= numTiles) return;            // uniform per wave -> EXEC all-ones inside
  int tm = (gwave / tilesN) << 4;
  int tn = (gwave % tilesN) << 4;
  int half = lane >> 4;                     // 0: lanes 0-15, 1: lanes 16-31
  int l    = lane & 15;

  v8f acc = {};
  const float* arow = A + (size_t)(tm + l) * K;
  for (int k0 = 0; k0 < K; k0 += 4) {
    int ka = k0 + (half << 1);
    v2f a, b;
    a.x = arow[ka];
    a.y = arow[ka + 1];
    b.x = B[(size_t)(ka)     * Ncol + tn + l];
    b.y = B[(size_t)(ka + 1) * Ncol + tn + l];
    acc = __builtin_amdgcn_wmma_f32_16x16x4_f32(false, a, false, b,
                                                (short)0, acc, false, false);
  }
  int row0 = tm + (half << 3);
  #pragma unroll
  for (int v = 0; v < 8; ++v)
    Cm[(size_t)(row0 + v) * Ncol + tn + l] = acc[v];
}

// ---------------- per-node attention scores ----------------
__global__ void attn_scores(const float* __restrict__ xw,
                            const float* __restrict__ a_src,
                            const float* __restrict__ a_dst,
                            float* __restrict__ esrc, float* __restrict__ edst,
                            int n, int H, int C) {
  int t = blockIdx.x * blockDim.x + threadIdx.x;
  if (t >= n * H) return;
  int node = t / H, h = t % H;
  const float* v  = xw + (size_t)node * H * C + (size_t)h * C;
  const float* as = a_src + (size_t)h * C;
  const float* ad = a_dst + (size_t)h * C;
  float ss = 0.f, sd = 0.f;
  for (int c = 0; c < C; ++c) { float x = v[c]; ss += x * as[c]; sd += x * ad[c]; }
  esrc[t] = ss;
  edst[t] = sd;
}

// ---------------- ordered-uint float max helpers ----------------
__device__ __forceinline__ unsigned ordered_enc(float f) {
  unsigned u = __float_as_uint(f);
  return (u & 0x80000000u) ? ~u : (u | 0x80000000u);
}
__device__ __forceinline__ float ordered_dec(unsigned o) {
  return (o & 0x80000000u) ? __uint_as_float(o & 0x7FFFFFFFu) : __uint_as_float(~o);
}

__device__ __forceinline__ void edge_nodes(const int* __restrict__ src,
                                           const int* __restrict__ dst,
                                           int e, int& s, int& d) {
  if (e < NEDGES) { s = src[e]; d = dst[e]; }
  else            { s = e - NEDGES; d = s; }   // self loops appended
}

// ---------------- edge pass A: logits + segment max ----------------
__global__ void edge_alpha_max(const int* __restrict__ src, const int* __restrict__ dst,
                               const float* __restrict__ esrc, const float* __restrict__ edst,
                               unsigned* __restrict__ amax, float* __restrict__ alpha, int H) {
  int t = blockIdx.x * blockDim.x + threadIdx.x;
  if (t >= ETOT * H) return;
  int e = t / H, h = t % H;
  int s, d; edge_nodes(src, dst, e, s, d);
  float a = esrc[s * H + h] + edst[d * H + h];
  a = (a >= 0.f) ? a : 0.2f * a;             // leaky_relu(0.2)
  alpha[t] = a;
  atomicMax(&amax[d * H + h], ordered_enc(a));
}

// ---------------- edge pass B: exp + segment sum ----------------
__global__ void edge_exp_denom(const int* __restrict__ src, const int* __restrict__ dst,
                               float* __restrict__ alpha_ex,
                               const unsigned* __restrict__ amax,
                               float* __restrict__ denom, int H) {
  int t = blockIdx.x * blockDim.x + threadIdx.x;
  if (t >= ETOT * H) return;
  int e = t / H, h = t % H;
  int s, d; edge_nodes(src, dst, e, s, d);
  float m  = ordered_dec(amax[d * H + h]);
  float ex = expf(alpha_ex[t] - m);
  alpha_ex[t] = ex;
  atomicAdd(&denom[d * H + h], ex);
}

// ---------------- edge pass C: weighted message scatter ----------------
__global__ void edge_aggregate(const int* __restrict__ src, const int* __restrict__ dst,
                               const float* __restrict__ xw,
                               const float* __restrict__ ex, const float* __restrict__ denom,
                               float* __restrict__ agg, int H, int C) {
  int F = H * C;
  int t = blockIdx.x * blockDim.x + threadIdx.x;
  if (t >= ETOT * F) return;
  int e = t / F, c = t % F;
  int h = c / C;
  int s, d; edge_nodes(src, dst, e, s, d);
  float a = ex[(size_t)e * H + h] / denom[d * H + h];
  atomicAdd(&agg[(size_t)d * F + c], xw[(size_t)s * F + c] * a);
}

// ---------------- bias (+ optional ELU), in place ----------------
__global__ void bias_act(float* __restrict__ x, const float* __restrict__ bias,
                         int n, int F, int doElu) {
  int t = blockIdx.x * blockDim.x + threadIdx.x;
  if (t >= n * F) return;
  int c = t % F;
  float v = x[t] + bias[c];
  if (doElu) v = (v > 0.f) ? v : (expf(v) - 1.f);
  x[t] = v;
}

// ---------------- global mean pool ----------------
__global__ void pool_count(const int* __restrict__ batch, float* __restrict__ cnt) {
  int n = blockIdx.x * blockDim.x + threadIdx.x;
  if (n >= NNODES) return;
  atomicAdd(&cnt[batch[n]], 1.f);
}
__global__ void pool_sum(const float* __restrict__ h, const int* __restrict__ batch,
                         float* __restrict__ out) {
  int t = blockIdx.x * blockDim.x + threadIdx.x;
  if (t >= NNODES * OUTD) return;
  int n = t / OUTD, c = t % OUTD;
  atomicAdd(&out[(size_t)batch[n] * OUTD + c], h[t]);
}
__global__ void pool_div(float* __restrict__ out, const float* __restrict__ cnt) {
  int t = blockIdx.x * blockDim.x + threadIdx.x;
  if (t >= NGRAPH * OUTD) return;
  out[t] /= fmaxf(cnt[t / OUTD], 1.f);
}

static inline int cdiv(long long a, int b) { return (int)((a + b - 1) / b); }

extern "C" void kernel_launch(void* const* d_in, const int* in_sizes, int n_in,
                              void* d_out, int out_size, void* d_ws, size_t ws_size,
                              hipStream_t stream) {
  (void)in_sizes; (void)n_in; (void)out_size; (void)ws_size;
  const float* x   = (const float*)d_in[0];
  const int*   ei  = (const int*)d_in[1];
  const int*   bat = (const int*)d_in[2];
  const float* W1  = (const float*)d_in[3];
  const float* as1 = (const float*)d_in[4];
  const float* ad1 = (const float*)d_in[5];
  const float* b1  = (const float*)d_in[6];
  const float* W2  = (const float*)d_in[7];
  const float* as2 = (const float*)d_in[8];
  const float* ad2 = (const float*)d_in[9];
  const float* b2  = (const float*)d_in[10];
  const int* src = ei;
  const int* dst = ei + NEDGES;

  // ---- workspace carve (layer 2 aliases layer 1's dead buffers) ----
  char* ws = (char*)d_ws;
  size_t off = 0;
  auto carve = [&](size_t bytes) -> char* {
    char* p = ws + off;
    off += (bytes + 255) & ~(size_t)255;
    return p;
  };
  float*    xw1    = (float*)carve((size_t)NNODES * HID * 4);
  float*    esrc1  = (float*)carve((size_t)NNODES * H1C * 4);
  float*    edst1  = (float*)carve((size_t)NNODES * H1C * 4);
  unsigned* amax1  = (unsigned*)carve((size_t)NNODES * H1C * 4);
  float*    denom1 = (float*)carve((size_t)NNODES * H1C * 4);
  float*    ex1    = (float*)carve((size_t)ETOT * H1C * 4);
  float*    agg1   = (float*)carve((size_t)NNODES * HID * 4);   // becomes layer-2 input h
  float*    agg2   = (float*)carve((size_t)NNODES * OUTD * 4);
  float*    cnt    = (float*)carve((size_t)NGRAPH * 4);
  // aliases (safe: used only after all layer-1 consumers have run on stream)
  float*    xw2    = xw1;
  float*    esrc2  = esrc1;
  float*    edst2  = edst1;
  unsigned* amax2  = amax1;
  float*    denom2 = denom1;
  float*    ex2    = ex1;
  float*    outF   = (float*)d_out;

  const int B = 256;

  // ---------------- layer 1: H=4, C=32 ----------------
  hipMemsetAsync(amax1,  0, (size_t)NNODES * H1C * 4, stream);
  hipMemsetAsync(denom1, 0, (size_t)NNODES * H1C * 4, stream);
  hipMemsetAsync(agg1,   0, (size_t)NNODES * HID * 4, stream);

  {
    int tiles = (NNODES / 16) * (HID / 16);
    gemm_wmma_f32<<<cdiv((long long)tiles * 32, B), B, 0, stream>>>(x, W1, xw1, NNODES, INDIM, HID);
  }
  attn_scores<<<cdiv((long long)NNODES * H1C, B), B, 0, stream>>>(xw1, as1, ad1, esrc1, edst1, NNODES, H1C, C1C);
  edge_alpha_max<<<cdiv((long long)ETOT * H1C, B), B, 0, stream>>>(src, dst, esrc1, edst1, amax1, ex1, H1C);
  edge_exp_denom<<<cdiv((long long)ETOT * H1C, B), B, 0, stream>>>(src, dst, ex1, amax1, denom1, H1C);
  edge_aggregate<<<cdiv((long long)ETOT * HID, B), B, 0, stream>>>(src, dst, xw1, ex1, denom1, agg1, H1C, C1C);
  bias_act<<<cdiv((long long)NNODES * HID, B), B, 0, stream>>>(agg1, b1, NNODES, HID, 1);

  // ---------------- layer 2: H=1, C=64 ----------------
  hipMemsetAsync(amax2,  0, (size_t)NNODES * 4, stream);
  hipMemsetAsync(denom2, 0, (size_t)NNODES * 4, stream);
  hipMemsetAsync(agg2,   0, (size_t)NNODES * OUTD * 4, stream);

  {
    int tiles = (NNODES / 16) * (OUTD / 16);
    gemm_wmma_f32<<<cdiv((long long)tiles * 32, B), B, 0, stream>>>(agg1, W2, xw2, NNODES, HID, OUTD);
  }
  attn_scores<<<cdiv((long long)NNODES, B), B, 0, stream>>>(xw2, as2, ad2, esrc2, edst2, NNODES, 1, OUTD);
  edge_alpha_max<<<cdiv((long long)ETOT, B), B, 0, stream>>>(src, dst, esrc2, edst2, amax2, ex2, 1);
  edge_exp_denom<<<cdiv((long long)ETOT, B), B, 0, stream>>>(src, dst, ex2, amax2, denom2, 1);
  edge_aggregate<<<cdiv((long long)ETOT * OUTD, B), B, 0, stream>>>(src, dst, xw2, ex2, denom2, agg2, 1, OUTD);
  bias_act<<<cdiv((long long)NNODES * OUTD, B), B, 0, stream>>>(agg2, b2, NNODES, OUTD, 0);

  // ---------------- global mean pool ----------------
  hipMemsetAsync(outF, 0, (size_t)NGRAPH * OUTD * 4, stream);
  hipMemsetAsync(cnt,  0, (size_t)NGRAPH * 4, stream);
  pool_count<<<cdiv(NNODES, B), B, 0, stream>>>(bat, cnt);
  pool_sum<<<cdiv((long long)NNODES * OUTD, B), B, 0, stream>>>(agg2, bat, outF);
  pool_div<<<cdiv((long long)NGRAPH * OUTD, B), B, 0, stream>>>(outF, cnt);
}